// RGCNConv_56023553409044
// MI455X (gfx1250) — compile-verified
//
#include <hip/hip_runtime.h>
#include <hip/hip_bf16.h>
#include <stdint.h>

typedef float    v2f  __attribute__((ext_vector_type(2)));
typedef float    v8f  __attribute__((ext_vector_type(8)));
typedef uint32_t u32x4 __attribute__((ext_vector_type(4)));
typedef uint32_t u32x8 __attribute__((ext_vector_type(8)));

// ---------------------------------------------------------------------------
// GEMM: H[n_rows x 128] = X[n_rows x 128] @ W[128 x 128], fp32 via
// V_WMMA_F32_16X16X4_F32. 128 threads = 4 waves, 64 rows / block.
//  - X tile staged LDS row-major (stride 132) by the Tensor Data Mover
//    (tensor_load_to_lds, padding done by TDM pad_interval/pad_amount).
//  - W staged per 32-K slab in fragment-major layout so every B fragment is
//    one aligned ds_load_b64 into an even VGPR pair (no repack movs).
// ---------------------------------------------------------------------------
#define KDIM   128
#define NDIM   128
#define TILE_M 64
#define XS     132     // padded LDS row stride (floats) for x tile
#define KSLAB  32      // K staged per slab (4 slabs)

__global__ __launch_bounds__(128)
void rgcn_gemm_f32(const float* __restrict__ X, const float* __restrict__ W,
                   float* __restrict__ H, int n_rows)
{
    __shared__ float sX[TILE_M * XS];        // 64 x 132 floats = 33,792 B
    __shared__ v2f   sWf[8 * 8 * 32];        // [k4][nb][lane] fragments, 16 KB

    const int tid  = threadIdx.x;
    const int wave = tid >> 5;               // 0..3
    const int lane = tid & 31;
    const int m    = lane & 15;              // M (or N) index within fragment
    const int kh   = lane >> 4;              // K half-select
    const int row0 = blockIdx.x * TILE_M;

    // ---- wave 0: issue TDM load of the 64x128 fp32 x-tile into sX ----------
    if (tid < 32) {
        uint64_t gaddr = (uint64_t)(uintptr_t)(X + (size_t)row0 * KDIM);
        uint32_t ldsa  = (uint32_t)(uintptr_t)&sX[0];   // low 32 bits = LDS offset
        uint32_t tdim1 = (uint32_t)(n_rows - row0);     // rows available (OOB -> 0-fill)

        u32x4 g0 = {
            1u,                                              // count=1, user mode
            ldsa,                                            // lds_addr
            (uint32_t)gaddr,                                 // global_addr[31:0]
            (uint32_t)((gaddr >> 32) & 0x01FFFFFFu) | (2u << 30) // [56:32] | type=2
        };
        u32x8 g1 = {
            (2u << 16) | (1u << 20) | (6u << 22) | (3u << 25), // 4B elems, pad: every 128 dw add 4 dw
            (128u & 0xFFFFu) << 16,                 // tensor_dim0[15:0]=128
            (tdim1 & 0xFFFFu) << 16,                // tensor_dim0[31:16]=0 | tensor_dim1[15:0]
            ((tdim1 >> 16) & 0xFFFFu) | (128u << 16), // tensor_dim1[31:16] | tile_dim0=128
            64u,                                    // tile_dim1=64, tile_dim2=0
            128u,                                   // tensor_dim0_stride[31:0]=128
            0u, 0u
        };
        u32x4 gz = {0u, 0u, 0u, 0u};                // groups 2/3 unused (2-D tile)

        asm volatile("tensor_load_to_lds %0, %1, %2, %3"
                     :: "s"(g0), "s"(g1), "s"(gz), "s"(gz)
                     : "memory");
        __builtin_amdgcn_s_wait_tensorcnt(0);
    }

    v8f acc[8];
    #pragma unroll
    for (int nb = 0; nb < 8; ++nb)
        acc[nb] = (v8f){0.f, 0.f, 0.f, 0.f, 0.f, 0.f, 0.f, 0.f};

    const int xrow = wave * 16 + m;          // this lane's A row in the tile

    for (int ks = 0; ks < KDIM / KSLAB; ++ks) {
        __syncthreads();                     // prev slab consumed / sX ready
        // ---- stage W k-slab into fragment-major layout --------------------
        #pragma unroll
        for (int i = 0; i < 8; ++i) {
            int idx = tid + i * 128;         // 1024 float4 slots (32 rows x 128)
            int r   = idx >> 5;              // local k row 0..31
            int c4  = (idx & 31) << 2;
            float4 v = *reinterpret_cast<const float4*>(
                           W + (size_t)(ks * KSLAB + r) * NDIM + c4);
            float vv[4] = {v.x, v.y, v.z, v.w};
            int k4  = r >> 2;
            int khw = (r >> 1) & 1;          // which half-wave owns this k row
            int s   = r & 1;                 // slot within the v2f fragment
            #pragma unroll
            for (int j = 0; j < 4; ++j) {
                int n = c4 + j;
                float* p = reinterpret_cast<float*>(
                    &sWf[(k4 * 8 + (n >> 4)) * 32 + khw * 16 + (n & 15)]);
                p[s] = vv[j];
            }
        }
        __syncthreads();

        #pragma unroll
        for (int k4 = 0; k4 < KSLAB / 4; ++k4) {
            // A fragment: 16x4 fp32 — two consecutive floats, 8B aligned
            const v2f a = *reinterpret_cast<const v2f*>(
                &sX[xrow * XS + ks * KSLAB + k4 * 4 + 2 * kh]);
            #pragma unroll
            for (int nb = 0; nb < 8; ++nb) {
                const v2f b = sWf[(k4 * 8 + nb) * 32 + lane]; // one ds_load_b64
                acc[nb] = __builtin_amdgcn_wmma_f32_16x16x4_f32(
                              false, a, false, b, (short)0, acc[nb],
                              false, false);
            }
        }
    }

    // ---- store D tiles: VGPR i -> (M = i + 8*kh, N = nb*16 + m) ------------
    #pragma unroll
    for (int nb = 0; nb < 8; ++nb) {
        #pragma unroll
        for (int i = 0; i < 8; ++i) {
            int grow = row0 + wave * 16 + i + 8 * kh;
            if (grow < n_rows)
                H[(size_t)grow * NDIM + nb * 16 + m] = acc[nb][i];
        }
    }
}

// ---------------------------------------------------------------------------
// Scatter: for edges with edge_type == rel, out[dst] += H[src] * norm.
// One wave32 per edge; lane handles 4 channels (float4 gather + 4 f32 atomics,
// both L2-resident working sets).
// ---------------------------------------------------------------------------
__global__ __launch_bounds__(256)
void rgcn_scatter(const float* __restrict__ H,
                  const int* __restrict__ edge_index,   // [2, E]
                  const int* __restrict__ edge_type,    // [E]
                  const float* __restrict__ edge_norm,  // [E]
                  float* __restrict__ out, int n_edges, int rel)
{
    const int tid    = blockIdx.x * blockDim.x + threadIdx.x;
    const int wid    = tid >> 5;
    const int lane   = threadIdx.x & 31;
    const int nwaves = (gridDim.x * blockDim.x) >> 5;

    for (int e = wid; e < n_edges; e += nwaves) {
        if (e + nwaves < n_edges)
            __builtin_prefetch(&edge_type[e + nwaves], 0, 1); // global_prefetch_b8
        if (edge_type[e] != rel) continue;
        const int src = edge_index[n_edges + e];   // edge_index[1] = source
        const int dst = edge_index[e];             // edge_index[0] = destination
        const float nrm = edge_norm[e];
        float4 h4 = reinterpret_cast<const float4*>(H + (size_t)src * 128)[lane];
        float* o = out + (size_t)dst * 128 + lane * 4;
        atomicAdd(o + 0, h4.x * nrm);
        atomicAdd(o + 1, h4.y * nrm);
        atomicAdd(o + 2, h4.z * nrm);
        atomicAdd(o + 3, h4.w * nrm);
    }
}

// ---------------------------------------------------------------------------
// Driver: out = x @ root; then for each relation r:
//   ws = x @ W[r]; scatter(type==r) into out.   (stream order = dependency)
// ---------------------------------------------------------------------------
extern "C" void kernel_launch(void* const* d_in, const int* in_sizes, int n_in,
                              void* d_out, int out_size, void* d_ws, size_t ws_size,
                              hipStream_t stream)
{
    const float* x          = (const float*)d_in[0];
    const float* weight_rel = (const float*)d_in[1];
    const float* root       = (const float*)d_in[2];
    const int*   edge_index = (const int*)d_in[3];
    const int*   edge_type  = (const int*)d_in[4];
    const float* edge_norm  = (const float*)d_in[5];

    const int n_nodes = in_sizes[0] / 128;
    const int n_rel   = in_sizes[1] / (128 * 128);
    const int n_edges = in_sizes[5];

    float* out = (float*)d_out;
    float* h   = (float*)d_ws;          // 25.6 MB scratch for one relation's h

    dim3 blk(128);
    dim3 grd((n_nodes + TILE_M - 1) / TILE_M);

    // out = x @ root  (plain store: also initializes poisoned d_out)
    rgcn_gemm_f32<<<grd, blk, 0, stream>>>(x, root, out, n_nodes);

    for (int r = 0; r < n_rel; ++r) {
        rgcn_gemm_f32<<<grd, blk, 0, stream>>>(
            x, weight_rel + (size_t)r * 128 * 128, h, n_nodes);
        rgcn_scatter<<<1024, 256, 0, stream>>>(
            h, edge_index, edge_type, edge_norm, out, n_edges, r);
    }
}